// DenseDilatedKnnGraph_26053271617654
// MI455X (gfx1250) — compile-verified
//
#include <hip/hip_runtime.h>

typedef __attribute__((ext_vector_type(2))) float v2f;
typedef __attribute__((ext_vector_type(8))) float v8f;

#define BATCH 2
#define CH    128
#define NPTS  8192
#define KCAND 18
#define KOUT  9
#define EPSN  1e-12f

// -------- Kernel 1: L2 normalize along channel dim, transpose to (B,N,C),
// and compute sq = sum(pts_normalized^2) per point (matches reference).
__global__ void knn_normalize(const float* __restrict__ x,
                              float* __restrict__ pts,
                              float* __restrict__ sq) {
  int p = blockIdx.x * blockDim.x + threadIdx.x;  // point id in [0, B*N)
  if (p >= BATCH * NPTS) return;
  int b = p / NPTS, n = p % NPTS;
  const float* xb = x + (size_t)b * CH * NPTS + n;  // x[b][c][n], c-stride = N
  float s = 0.f;
  for (int c = 0; c < CH; ++c) { float v = xb[(size_t)c * NPTS]; s += v * v; }
  float inv = 1.0f / fmaxf(sqrtf(s), EPSN);
  float s2 = 0.f;
  float* po = pts + (size_t)p * CH;
  for (int c = 0; c < CH; ++c) {
    float v = xb[(size_t)c * NPTS] * inv;
    po[c] = v;
    s2 += v * v;
  }
  sq[p] = s2;
}

// Monotone map: float total order -> u32 unsigned order (sign-flip trick).
__device__ __forceinline__ unsigned int fmap(float f) {
  unsigned int u = __float_as_uint(f);
  return (u & 0x80000000u) ? ~u : (u | 0x80000000u);
}
// Sortable key: (dist, idx) lexicographic == u64 unsigned compare.
// Matches lax.top_k(-dist) tie-break (smaller dist, then smaller index).
__device__ __forceinline__ unsigned long long mkkey(float d, int j) {
  return ((unsigned long long)fmap(d) << 32) | (unsigned int)j;
}

// Branchless sorted-list insert (ascending, slot 0 best). Caller guarantees
// key < kv[KCAND-1].
__device__ __forceinline__ void insert_key(unsigned long long* kv,
                                           unsigned long long key) {
  bool prev = true;
#pragma unroll
  for (int s = KCAND - 1; s > 0; --s) {
    bool sh = key < kv[s - 1];
    kv[s] = sh ? kv[s - 1] : (prev ? key : kv[s]);
    prev = sh;
  }
  if (prev) kv[0] = key;
}

// -------- Kernel 2: fused fp32-WMMA Gram matrix + streaming top-18 + dilation.
__launch_bounds__(256)
__global__ void knn_topk(const float* __restrict__ pts,
                         const float* __restrict__ sq,
                         int* __restrict__ out) {
  __shared__ float As[16][CH];                      // 8 KB: A block rows
  __shared__ float sqs[16];
  __shared__ float Ds[8][16][16];                   // 8 KB: per-wave dist tiles
  __shared__ unsigned long long Lk[16][16][KCAND];  // 36 KB: per-row candidates

  const int tid  = threadIdx.x;
  const int wave = tid >> 5;
  const int lane = tid & 31;
  const int b    = blockIdx.x / (NPTS / 16);
  const int row0 = (blockIdx.x % (NPTS / 16)) * 16;

  const float* P  = pts + (size_t)b * NPTS * CH;
  const float* SQ = sq  + (size_t)b * NPTS;

  // stage A block (16 rows x 128 ch) + row sq into LDS
  for (int i = tid; i < 16 * CH; i += 256) {
    int r = i / CH, c = i % CH;
    As[r][c] = P[(size_t)(row0 + r) * CH + c];
  }
  if (tid < 16) sqs[tid] = SQ[row0 + tid];
  __syncthreads();

  // per-lane sorted top-KCAND keys, register resident
  unsigned long long kv[KCAND];
#pragma unroll
  for (int s = 0; s < KCAND; ++s) kv[s] = ~0ull;

  const int m     = lane & 15;            // M (A rows) / N (B cols) within tile
  const int koff  = (lane < 16) ? 0 : 2;  // f32 16x16x4 lane-half K offset
  const int myrow = lane >> 1;            // transposed-read row ownership
  const int chalf = (lane & 1) * 8;       // transposed-read column half

  const int NT = NPTS / 16;               // 512 column tiles; 64 per wave
  for (int tile = wave; tile < NT; tile += 8) {
    const int col0 = tile * 16;
    v8f acc = {};
    const float* Bp = P + (size_t)(col0 + m) * CH;
#pragma unroll 4
    for (int ks = 0; ks < CH / 4; ++ks) {
      int k0 = ks * 4 + koff;
      v2f a  = *(const v2f*)&As[m][k0];   // A[M=m][K=k0..k0+1]
      v2f bb = *(const v2f*)&Bp[k0];      // B[K][N=m] = pts[col0+m][k0..k0+1]
      acc = __builtin_amdgcn_wmma_f32_16x16x4_f32(false, a, false, bb,
                                                  (short)0, acc, false, false);
    }
    // dist tile into LDS (C/D layout: lane holds rows rbase..rbase+7, col m)
    float sqj   = SQ[col0 + m];
    int   rbase = (lane >> 4) * 8;
#pragma unroll
    for (int v = 0; v < 8; ++v) {
      int r = rbase + v;
      Ds[wave][r][m] = sqs[r] + sqj - 2.0f * acc[v];
    }
    __syncthreads();
    // transposed read: lane owns (row = lane/2, 8 columns); j increasing
#pragma unroll
    for (int t = 0; t < 8; ++t) {
      int   c = chalf + t;
      float d = Ds[wave][myrow][c];
      unsigned long long key = mkkey(d, col0 + c);
      if (key < kv[KCAND - 1]) insert_key(kv, key);
    }
    __syncthreads();
  }

  // dump per-lane lists: row = lane/2 ; slot = wave*2 + (lane&1)
  {
    int slot = wave * 2 + (lane & 1);
#pragma unroll
    for (int s = 0; s < KCAND; ++s) Lk[myrow][slot][s] = kv[s];
  }
  __syncthreads();

  // merge 16 lists (288 candidates) per row; one thread per row
  if (tid < 16) {
    unsigned long long mv[KCAND];
#pragma unroll
    for (int s = 0; s < KCAND; ++s) mv[s] = ~0ull;
    for (int slot = 0; slot < 16; ++slot) {
      for (int e = 0; e < KCAND; ++e) {
        unsigned long long key = Lk[tid][slot][e];
        if (key < mv[KCAND - 1]) insert_key(mv, key);
      }
    }
    // dilation: every 2nd candidate; output (2, B, N, KOUT) int32
    int n = row0 + tid;
    size_t base  = ((size_t)b * NPTS + n) * KOUT;
    const size_t plane = (size_t)BATCH * NPTS * KOUT;
#pragma unroll
    for (int kk = 0; kk < KOUT; ++kk) {
      out[base + kk]         = (int)(mv[2 * kk] & 0xffffffffull);  // nn_idx
      out[plane + base + kk] = n;                                  // center_idx
    }
  }
}

extern "C" void kernel_launch(void* const* d_in, const int* in_sizes, int n_in,
                              void* d_out, int out_size, void* d_ws, size_t ws_size,
                              hipStream_t stream) {
  const float* x = (const float*)d_in[0];
  float* pts = (float*)d_ws;                          // B*N*CH floats (8 MB)
  float* sqb = pts + (size_t)BATCH * NPTS * CH;       // B*N floats
  int*   out = (int*)d_out;

  knn_normalize<<<(BATCH * NPTS + 255) / 256, 256, 0, stream>>>(x, pts, sqb);
  knn_topk<<<BATCH * (NPTS / 16), 256, 0, stream>>>(pts, sqb, out);
}